// multihead_attention_27650999451988
// MI455X (gfx1250) — compile-verified
//
#include <hip/hip_runtime.h>

typedef __attribute__((ext_vector_type(2))) float v2f;
typedef __attribute__((ext_vector_type(8))) float v8f;

#define T_DIM 128
#define C_DIM 384
#define H_DIM 24
#define SX_STRIDE 68    // x staging: 16B-aligned rows, conflict-free A reads
#define SK_STRIDE 25    // K: odd stride -> conflict-free
#define SQ_STRIDE 130   // Q^T rows
#define SV_STRIDE 40    // V rows: half-wave bank shift of 16 -> conflict-free
#define SS_STRIDE 132   // scores: bank stride 4 -> conflict-free A reads

__device__ __forceinline__ v8f wmma4(v2f a, v2f b, v8f c) {
    // D = A(16x4, f32) * B(4x16, f32) + C(16x16, f32)
    return __builtin_amdgcn_wmma_f32_16x16x4_f32(
        /*neg_a=*/false, a, /*neg_b=*/false, b,
        /*c_mod=*/(short)0, c, /*reuse_a=*/false, /*reuse_b=*/false);
}

__global__ __launch_bounds__(256, 1)
void attn_fused_kernel(const float* __restrict__ x,
                       const float* __restrict__ Wk,
                       const float* __restrict__ Wq,
                       const float* __restrict__ Wv,
                       float* __restrict__ out)
{
    // Scores buffer; its low 34816 B double as the x staging buffer (disjoint lifetimes).
    __shared__ float sS[T_DIM * SS_STRIDE];           // 67584 B
    __shared__ float sK[T_DIM * SK_STRIDE];           // 12800 B
    __shared__ float sQt[H_DIM * SQ_STRIDE];          // 12480 B
    __shared__ float sV[T_DIM * SV_STRIDE];           // 20480 B
    float* sX = sS;                                   // overlay

    const int b    = blockIdx.x;
    const int tid  = threadIdx.x;
    const int wave = tid >> 5;        // 0..7  -> M-tile
    const int lane = tid & 31;
    const int half = lane >> 4;       // 0/1
    const int n    = lane & 15;       // column-in-tile / M-in-tile
    const int kloc = half * 2;        // A/B fragment K offset within K=4 step

    const float* xb = x + (size_t)b * T_DIM * C_DIM;

    // ---------------- Phase 1: K/Q/V = x @ W ----------------
    v8f aK0 = {}, aK1 = {}, aQ0 = {}, aQ1 = {}, aV0 = {}, aV1 = {};
    const int  mt    = wave;
    const int  row_m = mt * 16 + n;        // A-fragment row (t)
    const int  nhi   = 16 + (n & 7);       // clamped tile-1 column (always < 24)
    const bool hi    = (n < 8);            // tile-1 validity mask (value select, no branch)

    for (int chunk = 0; chunk < C_DIM / 64; ++chunk) {
        const int c0 = chunk * 64;
        __syncthreads();
        // coalesced 128x64 chunk load: 2048 float4 over 256 threads
        for (int f = tid; f < T_DIM * 16; f += 256) {
            const int r  = f >> 4;
            const int cf = (f & 15) << 2;
            const float4 v = *(const float4*)(xb + (size_t)r * C_DIM + c0 + cf);
            *(float4*)(&sX[r * SX_STRIDE + cf]) = v;
        }
        if (chunk + 1 < C_DIM / 64) // warm next chunk (one 128B line per thread)
            __builtin_prefetch(xb + (size_t)(tid >> 1) * C_DIM + c0 + 64 + ((tid & 1) << 5), 0, 1);
        __syncthreads();

        for (int kk = 0; kk < 16; ++kk) {
            const int cl = kk * 4 + kloc;
            const v2f a  = *(const v2f*)(&sX[row_m * SX_STRIDE + cl]);
            const int c  = c0 + cl;
            const float* wk = Wk + (size_t)c * H_DIM;
            const float* wq = Wq + (size_t)c * H_DIM;
            const float* wv = Wv + (size_t)c * H_DIM;
            // Unconditional loads; tile-1 uses clamped column + value select.
            v2f bk0, bq0, bv0, bk1, bq1, bv1;
            bk0.x = wk[n];    bk0.y = wk[H_DIM + n];
            bq0.x = wq[n];    bq0.y = wq[H_DIM + n];
            bv0.x = wv[n];    bv0.y = wv[H_DIM + n];
            const float k1a = wk[nhi], k1b = wk[H_DIM + nhi];
            const float q1a = wq[nhi], q1b = wq[H_DIM + nhi];
            const float v1a = wv[nhi], v1b = wv[H_DIM + nhi];
            bk1.x = hi ? k1a : 0.f;  bk1.y = hi ? k1b : 0.f;
            bq1.x = hi ? q1a : 0.f;  bq1.y = hi ? q1b : 0.f;
            bv1.x = hi ? v1a : 0.f;  bv1.y = hi ? v1b : 0.f;

            aK0 = wmma4(a, bk0, aK0);  aK1 = wmma4(a, bk1, aK1);
            aQ0 = wmma4(a, bq0, aQ0);  aQ1 = wmma4(a, bq1, aQ1);
            aV0 = wmma4(a, bv0, aV0);  aV1 = wmma4(a, bv1, aV1);
        }
    }

    // write K [t][h], Q^T [h][t], V [s][h] (V padding cols 24..31 are true zeros)
    for (int j = 0; j < 8; ++j) {
        const int t = mt * 16 + half * 8 + j;
        sK[t * SK_STRIDE + n] = aK0[j];
        if (n < 8) sK[t * SK_STRIDE + 16 + n] = aK1[j];
        sQt[n * SQ_STRIDE + t] = aQ0[j];
        if (n < 8) sQt[(16 + n) * SQ_STRIDE + t] = aQ1[j];
        sV[t * SV_STRIDE + n]      = aV0[j];
        sV[t * SV_STRIDE + 16 + n] = aV1[j];
    }
    __syncthreads();   // sX reads done -> sS may be written

    // ---------------- Phase 2: S = K @ Q^T (causal tiles only) ----------------
    // A-fragments depend only on (mt, k): preload all 6 K-steps once, reuse per tile.
    v2f aS[H_DIM / 4];
    for (int kk = 0; kk < H_DIM / 4; ++kk) {
        const int k = kk * 4 + kloc;
        aS[kk].x = sK[(mt * 16 + n) * SK_STRIDE + k];
        aS[kk].y = sK[(mt * 16 + n) * SK_STRIDE + k + 1];
    }
    for (int nt = 0; nt <= mt; ++nt) {
        v8f acc = {};
        for (int kk = 0; kk < H_DIM / 4; ++kk) {
            const int k = kk * 4 + kloc;
            v2f bq;
            bq.x = sQt[k * SQ_STRIDE + nt * 16 + n];
            bq.y = sQt[(k + 1) * SQ_STRIDE + nt * 16 + n];
            acc  = wmma4(aS[kk], bq, acc);
        }
        for (int j = 0; j < 8; ++j)
            sS[(mt * 16 + half * 8 + j) * SS_STRIDE + nt * 16 + n] = acc[j];
    }
    __syncthreads();

    // ---------------- Phase 3: causal softmax (one thread per row) ----------------
    if (tid < T_DIM) {
        const int r = tid;
        float* rowp = &sS[r * SS_STRIDE];
        float mx = rowp[0];
        for (int s = 1; s <= r; ++s) mx = fmaxf(mx, rowp[s]);
        float sum = 0.f;
        for (int s = 0; s <= r; ++s) { const float e = __expf(rowp[s] - mx); rowp[s] = e; sum += e; }
        const float inv = 1.f / sum;
        for (int s = 0; s <= r; ++s) rowp[s] *= inv;
        for (int s = r + 1; s < T_DIM; ++s) rowp[s] = 0.f;   // masked tail -> exact zeros
    }
    __syncthreads();

    // ---------------- Phase 4: out = softmax(S) @ V ----------------
    // Interleave both N-tiles so each A-fragment read feeds two WMMAs.
    {
        v8f acc0 = {}, acc1 = {};
        for (int k0 = 0; k0 < T_DIM; k0 += 4) {
            const int k = k0 + kloc;
            v2f a, bv0, bv1;
            a.x   = sS[(mt * 16 + n) * SS_STRIDE + k];
            a.y   = sS[(mt * 16 + n) * SS_STRIDE + k + 1];
            bv0.x = sV[k * SV_STRIDE + n];
            bv0.y = sV[(k + 1) * SV_STRIDE + n];
            bv1.x = sV[k * SV_STRIDE + 16 + n];
            bv1.y = sV[(k + 1) * SV_STRIDE + 16 + n];
            acc0  = wmma4(a, bv0, acc0);
            acc1  = wmma4(a, bv1, acc1);
        }
        for (int j = 0; j < 8; ++j) {
            const int t = mt * 16 + half * 8 + j;
            out[((size_t)b * T_DIM + t) * H_DIM + n] = acc0[j];
            if (n < 8)
                out[((size_t)b * T_DIM + t) * H_DIM + 16 + n] = acc1[j];
        }
    }
}

extern "C" void kernel_launch(void* const* d_in, const int* in_sizes, int n_in,
                              void* d_out, int out_size, void* d_ws, size_t ws_size,
                              hipStream_t stream) {
    (void)n_in; (void)out_size; (void)d_ws; (void)ws_size;
    const float* x  = (const float*)d_in[0];
    const float* Wk = (const float*)d_in[1];
    const float* Wq = (const float*)d_in[2];
    const float* Wv = (const float*)d_in[3];
    float* out = (float*)d_out;
    const int Bn = in_sizes[0] / (T_DIM * C_DIM);   // 1024
    attn_fused_kernel<<<Bn, 256, 0, stream>>>(x, Wk, Wq, Wv, out);
}